// RNN_circular_ND_71081708749396
// MI455X (gfx1250) — compile-verified
//
#include <hip/hip_runtime.h>

// RNN recurrence h_{t+1} = h_t @ Wh^T + x_t @ Wx^T on gfx1250 (MI455X).
// 16 workgroups x 512 threads (16 wave32). Each WG owns 16 batch rows and
// runs the whole serial T loop with Wh resident in LDS; each wave owns a
// 16-wide column slice and accumulates one 16x16 f32 WMMA tile per step.
// K loop is software-pipelined (distance-8 register circular buffer) so
// every ds_load lands ~8 WMMAs before use.

typedef __attribute__((ext_vector_type(2))) float v2f;
typedef __attribute__((ext_vector_type(8))) float v8f;

#define B_   256
#define T_   1024
#define I_   8
#define H_   256
#define ROWS 16          // batch rows per workgroup
#define LDH  260         // padded LDS row stride in floats (bank-conflict free)
#define NTHREADS 512
#define KF   (H_ / 4)    // 64 K=4 fragments
#define PF   8           // software-pipeline prefetch distance

__global__ __launch_bounds__(NTHREADS)
void rnn_recurrence_wmma_f32(const float* __restrict__ x,    // (B,T,I)
                             const float* __restrict__ Wh,   // (H,H)
                             const float* __restrict__ Wx,   // (H,I)
                             const float* __restrict__ h0,   // (1,H)
                             float* __restrict__ out)        // (T+1,B,H)
{
    extern __shared__ float smem[];
    float* wh_lds = smem;                      // [H_][LDH]  = 266,240 B
    float* hbuf0  = smem + H_ * LDH;           // [ROWS][LDH]
    float* hbuf1  = hbuf0 + ROWS * LDH;        // [ROWS][LDH]

    const int tid   = threadIdx.x;
    const int lane  = tid & 31;
    const int wave  = tid >> 5;            // 0..15
    const int bm    = blockIdx.x * ROWS;   // batch row base of this WG
    const int cn    = wave * 16;           // output column base of this wave
    const int lrow  = lane & 15;           // 0..15
    const int lhalf = lane >> 4;           // 0 or 1

    // ---- stage Wh into LDS (row-major, padded rows), float4 copies ----
    for (int idx = tid; idx < (H_ * H_) / 4; idx += NTHREADS) {
        const int c  = idx >> 6;           // Wh row (output column)
        const int k4 = (idx & 63) << 2;    // K offset
        const float4 v = *(const float4*)(Wh + c * H_ + k4);
        *(float4*)(wh_lds + c * LDH + k4) = v;
    }

    // ---- init h_cur = broadcast h0; emit out[0] slice ----
    for (int idx = tid; idx < ROWS * H_; idx += NTHREADS) {
        const int row = idx >> 8;
        const int k   = idx & (H_ - 1);
        const float v = h0[k];
        hbuf0[row * LDH + k] = v;
        __builtin_nontemporal_store(v, out + (size_t)(bm + row) * H_ + k);
    }

    // ---- constant B fragments of Wx^T (K=8 split into two K=4 frags) ----
    // B frag layout: b[r] @ lane = Wx[cn+lrow][k0 + r + 2*lhalf]
    const float* wxrow = Wx + (cn + lrow) * I_ + 2 * lhalf;
    v2f wxf0; wxf0[0] = wxrow[0]; wxf0[1] = wxrow[1];
    v2f wxf1; wxf1[0] = wxrow[4]; wxf1[1] = wxrow[5];

    // ---- per-lane global x pointer (A frag rows of X_t, 16x8) ----
    const float* xptr = x + (size_t)(bm + lrow) * T_ * I_ + 2 * lhalf;

    // ---- per-lane LDS fragment offsets (float indices, += kk*4) ----
    const int haddr = lrow * LDH + 2 * lhalf;              // A: h rows
    const int waddr = (cn + lrow) * LDH + 2 * lhalf;       // B: Wh^T cols

    // ---- per-lane h writeback base pointers (one per buffer) ----
    // C layout: element r -> (row = r + 8*lhalf, col = cn + lrow)
    const int stoff = (8 * lhalf) * LDH + cn + lrow;
    float* st_cur = hbuf1 + stoff;   // store target while reading hbuf0
    float* st_alt = hbuf0 + stoff;

    __syncthreads();

    float* hc = hbuf0;               // buffer currently holding h_t

    // first timestep's x fragments
    v2f xa0; xa0[0] = xptr[0]; xa0[1] = xptr[1];
    v2f xa1; xa1[0] = xptr[4]; xa1[1] = xptr[5];

    for (int t = 0; t < T_; ++t) {
        // prefetch next timestep's x fragments (overlaps the K loop)
        v2f nx0 = xa0, nx1 = xa1;
        if (t + 1 < T_) {
            const float* xp = xptr + (size_t)(t + 1) * I_;
            nx0[0] = xp[0]; nx0[1] = xp[1];
            nx1[0] = xp[4]; nx1[1] = xp[5];
        }

        // C = X_t(16x8) @ Wx^T(8x16): two K=4 WMMAs seed the accumulator
        v8f c = {0.f, 0.f, 0.f, 0.f, 0.f, 0.f, 0.f, 0.f};
        c = __builtin_amdgcn_wmma_f32_16x16x4_f32(false, xa0, false, wxf0,
                                                  (short)0, c, false, false);
        c = __builtin_amdgcn_wmma_f32_16x16x4_f32(false, xa1, false, wxf1,
                                                  (short)0, c, false, false);

        // ---- C += h(16x256) @ Wh^T(256x16): 64 chained K=4 WMMAs,
        //      software pipelined with a distance-PF register ring ----
        v2f ha[PF], wb[PF];
        #pragma unroll
        for (int j = 0; j < PF; ++j) {
            ha[j] = *(const v2f*)(hc + haddr + j * 4);
            wb[j] = *(const v2f*)(wh_lds + waddr + j * 4);
        }
        #pragma unroll
        for (int kk = 0; kk < KF; ++kk) {
            const int s = kk & (PF - 1);
            const v2f a = ha[s];
            const v2f b = wb[s];
            if (kk + PF < KF) {
                ha[s] = *(const v2f*)(hc + haddr + (kk + PF) * 4);
                wb[s] = *(const v2f*)(wh_lds + waddr + (kk + PF) * 4);
            }
            c = __builtin_amdgcn_wmma_f32_16x16x4_f32(false, a, false, b,
                                                      (short)0, c, false, false);
        }

        // write h_new to LDS (immediate DS offsets) and stream to out[t+1]
        float* outt = out + (size_t)(t + 1) * (B_ * H_)
                          + (size_t)(bm + 8 * lhalf) * H_ + (cn + lrow);
        #pragma unroll
        for (int r = 0; r < 8; ++r) {
            const float v = c[r];
            st_cur[r * LDH] = v;                       // ds offset r*1040
            __builtin_nontemporal_store(v, outt + r * H_);
        }

        __syncthreads();            // publish h_new before next step reads it
        hc = (hc == hbuf0) ? hbuf1 : hbuf0;
        { float* tmp = st_cur; st_cur = st_alt; st_alt = tmp; }
        xa0 = nx0; xa1 = nx1;
    }
}

extern "C" void kernel_launch(void* const* d_in, const int* in_sizes, int n_in,
                              void* d_out, int out_size, void* d_ws, size_t ws_size,
                              hipStream_t stream) {
    const float* x  = (const float*)d_in[0];   // (B,T,I)
    const float* Wh = (const float*)d_in[1];   // (H,H)
    const float* Wx = (const float*)d_in[2];   // (H,I)
    const float* h0 = (const float*)d_in[3];   // (1,H)
    float* out = (float*)d_out;                // (T+1,B,H)

    const size_t lds_bytes = (size_t)(H_ * LDH + 2 * ROWS * LDH) * sizeof(float);
    hipLaunchKernelGGL(rnn_recurrence_wmma_f32,
                       dim3(B_ / ROWS), dim3(NTHREADS), lds_bytes, stream,
                       x, Wh, Wx, h0, out);
}